// RAGVT5_85014582657079
// MI455X (gfx1250) — compile-verified
//
#include <hip/hip_runtime.h>
#include <math.h>

typedef __attribute__((ext_vector_type(2))) float v2f;
typedef __attribute__((ext_vector_type(8))) float v8f;

#define DIM     768
#define TOKENS  32
#define QTOK    24
#define NCHUNK  512
#define BATCH   16
#define LSTRIDE 772   // 768 + 4: padded LDS row stride (floats) -> conflict-free WMMA A loads

__device__ __forceinline__ float wave_reduce_sum(float v) {
    #pragma unroll
    for (int off = 16; off > 0; off >>= 1)
        v += __shfl_down(v, off, 32);
    return v;
}

// ---------------- Kernel 1: question pooling (raw sum + norm; counts cancel) ---------
__global__ __launch_bounds__(192)
void qpool_kernel(const float* __restrict__ emb, const int* __restrict__ q_ids,
                  const int* __restrict__ q_mask, float* __restrict__ qsum,
                  float* __restrict__ qnorm) {
    const int b = blockIdx.x;
    const int f = threadIdx.x;               // float4 index 0..191
    float4 acc = make_float4(0.f, 0.f, 0.f, 0.f);
    for (int t = 0; t < QTOK; ++t) {
        if (q_mask[b * QTOK + t]) {
            const int id = q_ids[b * QTOK + t];
            const float4 r = ((const float4*)(emb + (size_t)id * DIM))[f];
            acc.x += r.x; acc.y += r.y; acc.z += r.z; acc.w += r.w;
        }
    }
    ((float4*)(qsum + b * DIM))[f] = acc;
    float sd = acc.x*acc.x + acc.y*acc.y + acc.z*acc.z + acc.w*acc.w;
    sd = wave_reduce_sum(sd);
    __shared__ float part[6];
    const int lane = threadIdx.x & 31, w = threadIdx.x >> 5;
    if (lane == 0) part[w] = sd;
    __syncthreads();
    if (threadIdx.x == 0) {
        float s = 0.f;
        #pragma unroll
        for (int i = 0; i < 6; ++i) s += part[i];
        qnorm[b] = sqrtf(s);
    }
}

// ------- Kernel 2: chunk pooling (gather, L2-resident) + WMMA cosine-sim GEMV --------
__global__ __launch_bounds__(512)
void chunk_sims_kernel(const float* __restrict__ emb,
                       const int* __restrict__ chunk_ids,
                       const int* __restrict__ chunk_mask,
                       const float* __restrict__ qsum,
                       const float* __restrict__ qnorm,
                       float* __restrict__ sims) {
    __shared__ float ldsS[16 * LSTRIDE];
    __shared__ float ldsNorm[16];

    const int lane  = threadIdx.x & 31;
    const int w     = threadIdx.x >> 5;          // wave id = chunk within 16-chunk tile
    const int tile  = blockIdx.x;                // 512 tiles total, 32 per batch
    const int b     = tile >> 5;
    const int cbase = (tile & 31) << 4;
    const int chunk = b * NCHUNK + cbase + w;

    // each lane owns one token's id/mask, broadcast via shfl (wave-uniform branches)
    const int id_l = chunk_ids [(size_t)chunk * TOKENS + lane];
    const int mk_l = chunk_mask[(size_t)chunk * TOKENS + lane];

    float4 acc[6];
    #pragma unroll
    for (int j = 0; j < 6; ++j) acc[j] = make_float4(0.f, 0.f, 0.f, 0.f);

    for (int t = 0; t < TOKENS; ++t) {
        const int mk = __shfl(mk_l, t, 32);
        if (mk) {
            const int id = __shfl(id_l, t, 32);
            const float4* row = (const float4*)(emb + (size_t)id * DIM);
            #pragma unroll
            for (int j = 0; j < 6; ++j) {       // coalesced global_load_b128
                const float4 r = row[lane + 32 * j];
                acc[j].x += r.x; acc[j].y += r.y; acc[j].z += r.z; acc[j].w += r.w;
            }
        }
    }

    // |S| for this chunk
    float sd = 0.f;
    #pragma unroll
    for (int j = 0; j < 6; ++j)
        sd += acc[j].x*acc[j].x + acc[j].y*acc[j].y + acc[j].z*acc[j].z + acc[j].w*acc[j].w;
    sd = wave_reduce_sum(sd);
    if (lane == 0) ldsNorm[w] = sqrtf(sd);

    // stash S tile in LDS (padded stride)
    #pragma unroll
    for (int j = 0; j < 6; ++j)
        *(float4*)&ldsS[w * LSTRIDE + 4 * (lane + 32 * j)] = acc[j];
    __syncthreads();

    if (w == 0) {
        // V_WMMA_F32_16X16X4_F32: D = A x B + C
        // A: 16 chunks x 4 dims from LDS (lane M=L%16..: VGPR0/1 hold K pair, halves K0/1 vs K2/3)
        // B: 4 x 16 with Q broadcast into all 16 columns -> column 0 is the GEMV result.
        const float* q = qsum + b * DIM;
        const int M    = lane & 15;
        const int koff = (lane >> 4) << 1;       // 0 (lanes 0-15) or 2 (lanes 16-31)
        v8f c = {};
        for (int kk = 0; kk < DIM; kk += 4) {
            v2f a  = *(const v2f*)&ldsS[M * LSTRIDE + kk + koff]; // ds_load_b64
            v2f bb = *(const v2f*)&q[kk + koff];
            c = __builtin_amdgcn_wmma_f32_16x16x4_f32(false, a, false, bb,
                                                      (short)0, c, false, false);
        }
        // C/D layout: lane 0 holds (M=0..7, N=0) in c[0..7]; lane 16 holds (M=8..15, N=0)
        const float qn = qnorm[b];
        if (lane == 0) {
            #pragma unroll
            for (int r = 0; r < 8; ++r)
                sims[b * NCHUNK + cbase + r] = c[r] / (ldsNorm[r] * qn);
        } else if (lane == 16) {
            #pragma unroll
            for (int r = 0; r < 8; ++r)
                sims[b * NCHUNK + cbase + 8 + r] = c[r] / (ldsNorm[8 + r] * qn);
        }
    }
}

// ---------------- Kernel 3: top-k (k from device scalar; stable lowest-index ties) ---
__global__ __launch_bounds__(512)
void topk_kernel(const float* __restrict__ sims, float* __restrict__ out,
                 const int* __restrict__ kptr) {
    __shared__ float sv[NCHUNK];
    __shared__ float rv[NCHUNK];
    __shared__ int   ri[NCHUNK];
    const int k = *kptr;
    const int b = blockIdx.x, tid = threadIdx.x;
    float* out_vals = out;
    int*   out_idx  = (int*)(out + BATCH * k);

    sv[tid] = sims[b * NCHUNK + tid];
    __syncthreads();
    for (int sel = 0; sel < k; ++sel) {
        rv[tid] = sv[tid]; ri[tid] = tid;
        __syncthreads();
        for (int s = NCHUNK / 2; s > 0; s >>= 1) {
            if (tid < s) {
                const float v1 = rv[tid], v2 = rv[tid + s];
                const int   i1 = ri[tid], i2 = ri[tid + s];
                if ((v2 > v1) || (v2 == v1 && i2 < i1)) { rv[tid] = v2; ri[tid] = i2; }
            }
            __syncthreads();
        }
        if (tid == 0) {
            out_vals[b * k + sel] = rv[0];
            out_idx [b * k + sel] = ri[0];
            sv[ri[0]] = -__builtin_inff();
        }
        __syncthreads();
    }
}

extern "C" void kernel_launch(void* const* d_in, const int* in_sizes, int n_in,
                              void* d_out, int out_size, void* d_ws, size_t ws_size,
                              hipStream_t stream) {
    const float* emb        = (const float*)d_in[0];   // [32128, 768] f32
    const int*   chunk_ids  = (const int*)  d_in[1];   // [16, 512, 32]
    const int*   chunk_mask = (const int*)  d_in[2];   // [16, 512, 32]
    const int*   q_ids      = (const int*)  d_in[3];   // [16, 24]
    const int*   q_mask     = (const int*)  d_in[4];   // [16, 24]
    const int*   kptr       = (const int*)  d_in[5];   // scalar k (=5)

    float* ws    = (float*)d_ws;
    float* qsum  = ws;                                  // 16*768
    float* qnorm = ws + BATCH * DIM;                    // 16
    float* sims  = qnorm + BATCH;                       // 16*512

    qpool_kernel<<<BATCH, 192, 0, stream>>>(emb, q_ids, q_mask, qsum, qnorm);
    chunk_sims_kernel<<<(BATCH * NCHUNK) / 16, 512, 0, stream>>>(
        emb, chunk_ids, chunk_mask, qsum, qnorm, sims);
    topk_kernel<<<BATCH, NCHUNK, 0, stream>>>(sims, (float*)d_out, kptr);
}